// GausSplatingHead_4088808866444
// MI455X (gfx1250) — compile-verified
//
#include <hip/hip_runtime.h>
#include <hip/hip_bf16.h>

typedef float v4f __attribute__((ext_vector_type(4)));

// Per-gaussian: cov = R * diag(s^2) * R^T, upper-triangular 6 elements.
__device__ __forceinline__ void cov6(float r, float x, float y, float z,
                                     float sx, float sy, float sz,
                                     float* __restrict__ c) {
    float n   = r * r + x * x + y * y + z * z;
    float inv = 1.0f / sqrtf(n);
    r *= inv; x *= inv; y *= inv; z *= inv;

    float R00 = 1.0f - 2.0f * (y * y + z * z);
    float R01 = 2.0f * (x * y - r * z);
    float R02 = 2.0f * (x * z + r * y);
    float R10 = 2.0f * (x * y + r * z);
    float R11 = 1.0f - 2.0f * (x * x + z * z);
    float R12 = 2.0f * (y * z - r * x);
    float R20 = 2.0f * (x * z - r * y);
    float R21 = 2.0f * (y * z + r * x);
    float R22 = 1.0f - 2.0f * (x * x + y * y);

    float vx = sx * sx, vy = sy * sy, vz = sz * sz;

    c[0] = R00 * R00 * vx + R01 * R01 * vy + R02 * R02 * vz; // c00
    c[1] = R00 * R10 * vx + R01 * R11 * vy + R02 * R12 * vz; // c01
    c[2] = R00 * R20 * vx + R01 * R21 * vy + R02 * R22 * vz; // c02
    c[3] = R10 * R10 * vx + R11 * R11 * vy + R12 * R12 * vz; // c11
    c[4] = R10 * R20 * vx + R11 * R21 * vy + R12 * R22 * vz; // c12
    c[5] = R20 * R20 * vx + R21 * R21 * vy + R22 * R22 * vz; // c22
}

// Main kernel: 4 gaussians per lane, all-b128 non-temporal memory traffic.
//   scales: 12 floats  -> 3 x global_load_b128  (th:NT)
//   rots:   16 floats  -> 4 x global_load_b128  (th:NT)
//   out:    24 floats  -> 6 x global_store_b128 (th:NT)
__global__ __launch_bounds__(256) void gs_cov4(const float* __restrict__ scales,
                                               const float* __restrict__ rots,
                                               float* __restrict__ out,
                                               int n4) {
    int i = blockIdx.x * blockDim.x + threadIdx.x;
    if (i >= n4) return;

    const v4f* sp = (const v4f*)scales;
    const v4f* rp = (const v4f*)rots;
    v4f*       op = (v4f*)out;

    size_t sbase = (size_t)i * 3;
    size_t rbase = (size_t)i * 4;
    size_t obase = (size_t)i * 6;

    // Speculative gfx1250 prefetch (global_prefetch_b8) a fixed stream-distance
    // ahead; dropped silently if the address is past the buffer.
    __builtin_prefetch((const char*)(rp + rbase) + (1 << 16), 0, 1);
    __builtin_prefetch((const char*)(sp + sbase) + (3 << 14), 0, 1);

    v4f s0 = __builtin_nontemporal_load(sp + sbase + 0);
    v4f s1 = __builtin_nontemporal_load(sp + sbase + 1);
    v4f s2 = __builtin_nontemporal_load(sp + sbase + 2);

    v4f q0 = __builtin_nontemporal_load(rp + rbase + 0);
    v4f q1 = __builtin_nontemporal_load(rp + rbase + 1);
    v4f q2 = __builtin_nontemporal_load(rp + rbase + 2);
    v4f q3 = __builtin_nontemporal_load(rp + rbase + 3);

    float c[24];
    cov6(q0[0], q0[1], q0[2], q0[3], s0[0], s0[1], s0[2], c + 0);
    cov6(q1[0], q1[1], q1[2], q1[3], s0[3], s1[0], s1[1], c + 6);
    cov6(q2[0], q2[1], q2[2], q2[3], s1[2], s1[3], s2[0], c + 12);
    cov6(q3[0], q3[1], q3[2], q3[3], s2[1], s2[2], s2[3], c + 18);

#pragma unroll
    for (int k = 0; k < 6; ++k) {
        v4f o;
        o[0] = c[k * 4 + 0];
        o[1] = c[k * 4 + 1];
        o[2] = c[k * 4 + 2];
        o[3] = c[k * 4 + 3];
        __builtin_nontemporal_store(o, op + obase + k);
    }
}

// Scalar tail kernel for N % 4 != 0 (not launched when N is a multiple of 4).
__global__ __launch_bounds__(256) void gs_cov1(const float* __restrict__ scales,
                                               const float* __restrict__ rots,
                                               float* __restrict__ out,
                                               int start, int n) {
    int i = start + blockIdx.x * blockDim.x + threadIdx.x;
    if (i >= n) return;

    float c[6];
    cov6(rots[(size_t)i * 4 + 0], rots[(size_t)i * 4 + 1],
         rots[(size_t)i * 4 + 2], rots[(size_t)i * 4 + 3],
         scales[(size_t)i * 3 + 0], scales[(size_t)i * 3 + 1],
         scales[(size_t)i * 3 + 2], c);

#pragma unroll
    for (int k = 0; k < 6; ++k)
        __builtin_nontemporal_store(c[k], out + (size_t)i * 6 + k);
}

extern "C" void kernel_launch(void* const* d_in, const int* in_sizes, int n_in,
                              void* d_out, int out_size, void* d_ws, size_t ws_size,
                              hipStream_t stream) {
    (void)n_in; (void)out_size; (void)d_ws; (void)ws_size;

    const float* scales = (const float*)d_in[0]; // (N, 3) f32
    const float* rots   = (const float*)d_in[1]; // (N, 4) f32
    float*       out    = (float*)d_out;         // (N, 6) f32

    int N   = in_sizes[1] / 4; // rots flat count / 4
    int n4  = N / 4;
    int rem = N - n4 * 4;

    if (n4 > 0) {
        int blocks = (n4 + 255) / 256;
        gs_cov4<<<blocks, 256, 0, stream>>>(scales, rots, out, n4);
    }
    if (rem > 0) {
        gs_cov1<<<1, 256, 0, stream>>>(scales, rots, out, n4 * 4, N);
    }
}